// BiMapGeo_36661840838886
// MI455X (gfx1250) — compile-verified
//
#include <hip/hip_runtime.h>

// BiMapGeo forward for MI455X (gfx1250, wave32).
// P[b,o] = sum_c W[o,c]^T @ X[b,c] @ W[o,c]
// fp32 WMMA (v_wmma_f32_16x16x4_f32) + TDM (tensor_load_to_lds) double-buffered
// staging of the X tile, with TDM padding generating the 132-dword LDS pitch.

typedef __attribute__((ext_vector_type(2))) float v2f;
typedef __attribute__((ext_vector_type(8))) float v8f;
typedef __attribute__((ext_vector_type(4))) unsigned int u32x4;
typedef __attribute__((ext_vector_type(4))) int i32x4;
typedef __attribute__((ext_vector_type(8))) int i32x8;

#define B_   256
#define HI_  8
#define HO_  16
#define NI_  128
#define NO_  64
#define PITCH 132   // LDS row pitch in dwords: even (b64 align), 16B-aligned rows, bank-conflict-free

// Issue one TDM descriptor copying a contiguous 128x128 f32 tile (64KB) from
// global memory into LDS at lds_byte_off, inserting 4 pad dwords after every
// 128 dwords (=> 132-dword row pitch). Tracked by TENSORcnt.
__device__ __forceinline__ void tdm_load_x_tile(const float* gsrc, unsigned lds_byte_off) {
  unsigned long long ga = (unsigned long long)(uintptr_t)gsrc;
  u32x4 g0;
  g0[0] = 1u;                                   // count=1 (valid user descriptor)
  g0[1] = lds_byte_off;                         // lds_addr (bytes)
  g0[2] = (unsigned)(ga & 0xFFFFFFFFu);         // global_addr[31:0]
  g0[3] = (unsigned)((ga >> 32) & 0x01FFFFFFu)  // global_addr[56:32]
        | (2u << 30);                           // type=2 ("image")
  i32x8 g1;
  g1[0] = (int)((2u << 16)    // data_size = 4 bytes
              | (1u << 20)    // pad_enable
              | (6u << 22)    // pad_interval: every 2^(6+1)=128 dwords
              | (3u << 25));  // pad_amount: 3+1 = 4 dwords
  g1[1] = (int)0x40000000;    // tensor_dim0[15:0]=16384 in bits[63:48]
  g1[2] = (int)0x00010000;    // tensor_dim0[31:16]=0, tensor_dim1[15:0]=1
  g1[3] = (int)0x40000000;    // tensor_dim1[31:16]=0, tile_dim0=16384
  g1[4] = 1;                  // tile_dim1=1, tile_dim2=0
  g1[5] = 16384;              // tensor_dim0_stride[31:0]
  g1[6] = (int)0x40000000;    // stride0[47:32]=0, tensor_dim1_stride[15:0]=16384
  g1[7] = 0;
  i32x4 z4 = {0, 0, 0, 0};    // groups 2/3 unused (<=2D tensor)
#if __has_include(<hip/amd_detail/amd_gfx1250_TDM.h>)
  i32x8 z8 = {0, 0, 0, 0, 0, 0, 0, 0};
  __builtin_amdgcn_tensor_load_to_lds(g0, g1, z4, z4, z8, 0);
#else
  __builtin_amdgcn_tensor_load_to_lds(g0, g1, z4, z4, 0);
#endif
}

__global__ __launch_bounds__(256)
void bimapgeo_wmma_kernel(const float* __restrict__ X,
                          const float* __restrict__ W,
                          float* __restrict__ P) {
  // LDS: double-buffered Xs[2][128][132] (filled by TDM with padding),
  // Wt[64][132] (W transposed: Wt[q][j] = W[j][q]), Tt[64][132] (T transposed).
  __shared__ float Xs[2][NI_ * PITCH];
  __shared__ float Wt[NO_ * PITCH];
  __shared__ float Tt[NO_ * PITCH];

  const int t    = threadIdx.x;     // 0..255, 8 waves
  const int lane = t & 31;
  const int wv   = t >> 5;          // wave id 0..7
  const int lm   = lane & 15;       // row/col within fragment
  const int lh   = lane >> 4;       // half-wave selector (K split for A/B frags)
  const int b    = blockIdx.x >> 4; // 0..255
  const int o    = blockIdx.x & 15; // 0..15

  const float* xbase0 = X + (size_t)(b * HI_) * (NI_ * NI_);

  // Stage-2 accumulators: wave wv owns tiles (pi, q0) and (pi, q0+1) of the
  // 4x4 tile grid of P (64x64). Persistent across the channel loop.
  const int pi = wv >> 1;
  const int q0 = (wv & 1) * 2;
  v8f accP0 = {};
  v8f accP1 = {};

  // Prologue: kick off the DMA for channel 0 into buffer 0.
  if (wv == 0) {
    tdm_load_x_tile(xbase0, (unsigned)(uintptr_t)&Xs[0][0]);
  }

  for (int c = 0; c < HI_; ++c) {
    const int buf = c & 1;
    __syncthreads();  // prior iteration done reading Wt/Tt and Xs[buf^1]

    // ---- Pipeline: issue DMA for channel c+1, then wait for channel c. ----
    if (wv == 0) {
      if (c + 1 < HI_) {
        tdm_load_x_tile(xbase0 + (size_t)(c + 1) * (NI_ * NI_),
                        (unsigned)(uintptr_t)&Xs[buf ^ 1][0]);
        __builtin_amdgcn_s_wait_tensorcnt(1);  // in-order: tile c has landed
      } else {
        __builtin_amdgcn_s_wait_tensorcnt(0);
      }
    }
    // ---- Load W[o,c] (128x64) transposed into Wt (coalesced read) ----
    {
      const float* wbase = W + (size_t)(o * HI_ + c) * (NI_ * NO_);
#pragma unroll
      for (int i = 0; i < 32; ++i) {
        const int idx = i * 256 + t;
        const int j = idx >> 6;   // 0..127
        const int q = idx & 63;   // 0..63
        Wt[q * PITCH + j] = wbase[idx];
      }
    }
    __syncthreads();  // Xs[buf] (TDM) and Wt visible to all waves

    // ---- Stage 1: T = Xs @ W. Wave wv computes T rows [16wv, 16wv+16). ----
    // A frag: Xs[buf][16wv+lm][k0 + 2*lh + {0,1}]  (b64)
    // B frag: W[k][n] = Wt[n][k] -> Wt[ni*16+lm][k0 + 2*lh + {0,1}] (b64)
    {
      v8f acc0 = {}, acc1 = {}, acc2 = {}, acc3 = {};
      const float* Arow = &Xs[buf][(16 * wv + lm) * PITCH + 2 * lh];
      const float* Bn0  = &Wt[(0 * 16 + lm) * PITCH + 2 * lh];
      const float* Bn1  = &Wt[(1 * 16 + lm) * PITCH + 2 * lh];
      const float* Bn2  = &Wt[(2 * 16 + lm) * PITCH + 2 * lh];
      const float* Bn3  = &Wt[(3 * 16 + lm) * PITCH + 2 * lh];
#pragma unroll 8
      for (int k = 0; k < NI_; k += 4) {
        v2f a  = *(const v2f*)(Arow + k);
        v2f b0 = *(const v2f*)(Bn0 + k);
        v2f b1 = *(const v2f*)(Bn1 + k);
        v2f b2 = *(const v2f*)(Bn2 + k);
        v2f b3 = *(const v2f*)(Bn3 + k);
        acc0 = __builtin_amdgcn_wmma_f32_16x16x4_f32(false, a, false, b0, (short)0, acc0, false, false);
        acc1 = __builtin_amdgcn_wmma_f32_16x16x4_f32(false, a, false, b1, (short)0, acc1, false, false);
        acc2 = __builtin_amdgcn_wmma_f32_16x16x4_f32(false, a, false, b2, (short)0, acc2, false, false);
        acc3 = __builtin_amdgcn_wmma_f32_16x16x4_f32(false, a, false, b3, (short)0, acc3, false, false);
      }
      // Write T transposed into LDS: D element (vgpr r, lane) is
      // T[m][n], m = 16wv + 8*lh + r, n = ni*16 + lm -> Tt[n][m], 8 contiguous floats.
      const int mbase = 16 * wv + 8 * lh;
      float4* d0 = (float4*)&Tt[(0 * 16 + lm) * PITCH + mbase];
      d0[0] = make_float4(acc0[0], acc0[1], acc0[2], acc0[3]);
      d0[1] = make_float4(acc0[4], acc0[5], acc0[6], acc0[7]);
      float4* d1 = (float4*)&Tt[(1 * 16 + lm) * PITCH + mbase];
      d1[0] = make_float4(acc1[0], acc1[1], acc1[2], acc1[3]);
      d1[1] = make_float4(acc1[4], acc1[5], acc1[6], acc1[7]);
      float4* d2 = (float4*)&Tt[(2 * 16 + lm) * PITCH + mbase];
      d2[0] = make_float4(acc2[0], acc2[1], acc2[2], acc2[3]);
      d2[1] = make_float4(acc2[4], acc2[5], acc2[6], acc2[7]);
      float4* d3 = (float4*)&Tt[(3 * 16 + lm) * PITCH + mbase];
      d3[0] = make_float4(acc3[0], acc3[1], acc3[2], acc3[3]);
      d3[1] = make_float4(acc3[4], acc3[5], acc3[6], acc3[7]);
    }
    __syncthreads();  // Tt complete

    // ---- Stage 2: P += W^T @ T.  A[p][i] = Wt[p][i], B[i][q] = Tt[q][i]. ----
    {
      const float* A2  = &Wt[(pi * 16 + lm) * PITCH + 2 * lh];
      const float* Bq0 = &Tt[((q0 + 0) * 16 + lm) * PITCH + 2 * lh];
      const float* Bq1 = &Tt[((q0 + 1) * 16 + lm) * PITCH + 2 * lh];
#pragma unroll 8
      for (int k = 0; k < NI_; k += 4) {
        v2f a  = *(const v2f*)(A2 + k);
        v2f b0 = *(const v2f*)(Bq0 + k);
        v2f b1 = *(const v2f*)(Bq1 + k);
        accP0 = __builtin_amdgcn_wmma_f32_16x16x4_f32(false, a, false, b0, (short)0, accP0, false, false);
        accP1 = __builtin_amdgcn_wmma_f32_16x16x4_f32(false, a, false, b1, (short)0, accP1, false, false);
      }
    }
  }

  // ---- Write P[b,o] (64x64). D element (r, lane): p = pi*16 + 8*lh + r,
  //      q = (q0+qq)*16 + lm. Coalesced 64B segments per half-wave. ----
  float* out = P + (size_t)(b * HO_ + o) * (NO_ * NO_);
  const int prow = pi * 16 + 8 * lh;
#pragma unroll
  for (int r = 0; r < 8; ++r) {
    out[(prow + r) * NO_ + (q0 + 0) * 16 + lm] = accP0[r];
    out[(prow + r) * NO_ + (q0 + 1) * 16 + lm] = accP1[r];
  }
}

extern "C" void kernel_launch(void* const* d_in, const int* in_sizes, int n_in,
                              void* d_out, int out_size, void* d_ws, size_t ws_size,
                              hipStream_t stream) {
  const float* X = (const float*)d_in[0];  // (256, 8, 128, 128) f32
  const float* W = (const float*)d_in[1];  // (16, 8, 128, 64)  f32
  float* P = (float*)d_out;                // (256, 16, 64, 64) f32
  (void)in_sizes; (void)n_in; (void)out_size; (void)d_ws; (void)ws_size;

  dim3 grid(B_ * HO_);  // 4096 workgroups: one per (b, o)
  dim3 block(256);      // 8 waves (wave32)
  hipLaunchKernelGGL(bimapgeo_wmma_kernel, grid, block, 0, stream, X, W, P);
}